// LightweightConv2d_10849087390387
// MI455X (gfx1250) — compile-verified
//
#include <hip/hip_runtime.h>
#include <hip/hip_bf16.h>
#include <stdint.h>

// Problem constants (from reference): B=16, C=128, H=224, W=224, 16 heads, K=7, pad=3
#define NUM_HEADS 16
#define KW 7
#define PADW 3
#define WDIM 224
#define NROWS (16 * 128 * 224)            // B*C*H = 458752 independent rows
#define LPAD 4                            // zeroed halo floats on each side of an LDS row
#define LROW (LPAD + WDIM + LPAD)         // 232 floats per padded LDS row
#define ROWS_PER_WAVE 8                   // one TDM tile (row) per iteration, double buffered
#define WAVES_PER_BLOCK 8
#define ROWS_PER_BLOCK (ROWS_PER_WAVE * WAVES_PER_BLOCK)   // 64
#define NBLOCKS (NROWS / ROWS_PER_BLOCK)                   // 7168 exactly

typedef unsigned int uint32;
typedef uint32 v4u __attribute__((ext_vector_type(4)));
typedef int    v4i __attribute__((ext_vector_type(4)));
typedef int    v8i __attribute__((ext_vector_type(8)));

#if defined(__has_builtin)
#  if __has_builtin(__builtin_amdgcn_tensor_load_to_lds) && __has_builtin(__builtin_amdgcn_s_wait_tensorcnt)
#    define USE_TDM 1
#  endif
#endif
#ifndef USE_TDM
#  define USE_TDM 0
#endif

#if USE_TDM
// Issue one TDM 1-D row load: 224 f32 from global address gaddr_bytes into LDS byte
// offset lds_addr. D# packing per CDNA5 ISA.
__device__ __forceinline__ void tdm_load_row(uint32 lds_addr, uint64_t gaddr_bytes)
{
    v4u g0;
    g0.x = 1u;                                          // count=1, user descriptor, no gather
    g0.y = lds_addr;                                    // lds_addr -> bits 63:32
    g0.z = (uint32)(gaddr_bytes & 0xFFFFFFFFull);       // global_addr[31:0]
    g0.w = (uint32)(gaddr_bytes >> 32) | 0x80000000u;   // global_addr[56:32] | type=2 at [127:126]

    v8i g1;
    g1[0] = 0x00020000;                                 // data_size=2 (4-byte elements); mask=0
    g1[1] = (int)((uint32)WDIM << 16);                  // tensor_dim0[15:0] -> bits 63:48
    g1[2] = (int)(1u << 16);                            // tensor_dim1[15:0] = 1 -> bits 95:80
    g1[3] = (int)((uint32)WDIM << 16);                  // tensor_dim1 hi=0 | tile_dim0=224 -> bits 127:112
    g1[4] = 1;                                          // tile_dim1 = 1 row
    g1[5] = WDIM;                                       // tensor_dim0_stride[31:0]
    g1[6] = 0;
    g1[7] = 0;

    v4i z4 = {0, 0, 0, 0};
#if defined(__clang_major__) && (__clang_major__ >= 23)
    v8i z8 = {0, 0, 0, 0, 0, 0, 0, 0};
    __builtin_amdgcn_tensor_load_to_lds(g0, g1, z4, z4, z8, 0);
#else
    __builtin_amdgcn_tensor_load_to_lds(g0, g1, z4, z4, 0);
#endif
}
#endif

__global__ __launch_bounds__(256) void
lwconv2d_kernel(const float* __restrict__ x,
                const float* __restrict__ weight,
                float* __restrict__ out)
{
    __shared__ float wsm[NUM_HEADS * KW];                              // softmaxed taps
    __shared__ __align__(16) float tiles[WAVES_PER_BLOCK][2][LROW];    // 14,848 B double-buffered

    const int tid  = threadIdx.x;
    const int wave = tid >> 5;
    const int lane = tid & 31;

    const int wave_row0 = blockIdx.x * ROWS_PER_BLOCK + wave * ROWS_PER_WAVE;

#if USE_TDM
    // Kick off the first row DMA immediately (per-wave; TDM ignores EXEC).
    {
        uint32 lds0 = (uint32)(size_t)(&tiles[wave][0][LPAD]);  // low 32 bits of flat ptr = LDS offset
        uint64_t g = (uint64_t)(size_t)x + (uint64_t)wave_row0 * (WDIM * 4);
        tdm_load_row(lds0, g);
    }
#endif

    // Zero the halo pads of both buffers (disjoint from the DMA's interior writes).
    if (lane < 16) {
        const int b = lane >> 3;                         // buffer 0/1
        const int p = lane & 7;                          // pad slot 0..7
        const int off = (p < LPAD) ? p : (LPAD + WDIM + (p - LPAD));
        tiles[wave][b][off] = 0.f;
    }

    // Softmax the 16x7 weight table once per block, overlapped with the DMA.
    if (tid < NUM_HEADS) {
        float wv[KW];
        float mx = -1e30f;
#pragma unroll
        for (int k = 0; k < KW; ++k) { wv[k] = weight[tid * KW + k]; mx = fmaxf(mx, wv[k]); }
        float s = 0.f;
#pragma unroll
        for (int k = 0; k < KW; ++k) { wv[k] = __expf(wv[k] - mx); s += wv[k]; }
        const float inv = 1.f / s;
#pragma unroll
        for (int k = 0; k < KW; ++k) wsm[tid * KW + k] = wv[k] * inv;
    }
    __syncthreads();

#pragma unroll
    for (int it = 0; it < ROWS_PER_WAVE; ++it) {
        const int cur = it & 1;

#if USE_TDM
        if (it + 1 < ROWS_PER_WAVE) {
            // Prefetch next row into the other buffer, then wait for current (in-order).
            uint32 ldsn = (uint32)(size_t)(&tiles[wave][(it + 1) & 1][LPAD]);
            uint64_t g = (uint64_t)(size_t)x + (uint64_t)(wave_row0 + it + 1) * (WDIM * 4);
            tdm_load_row(ldsn, g);
            __builtin_amdgcn_s_wait_tensorcnt(1);
        } else {
            __builtin_amdgcn_s_wait_tensorcnt(0);
        }
        asm volatile("" ::: "memory");   // keep LDS reads below the tensorcnt wait
#else
        // Fallback staging: wave copies its row (56 float4 chunks) into the padded slot.
        {
            const float4* src = (const float4*)(x + (size_t)(wave_row0 + it) * WDIM);
            float4* dst = (float4*)(&tiles[wave][cur][LPAD]);
#pragma unroll
            for (int j = 0; j < 2; ++j) {
                int idx = lane + j * 32;
                if (idx < WDIM / 4) dst[idx] = src[idx];
            }
        }
#endif

        const int r = wave_row0 + it;
        const int head = (r / 224) & (NUM_HEADS - 1);   // (r/H) % 16 == channel % 16

        float wk[KW];
#pragma unroll
        for (int k = 0; k < KW; ++k) wk[k] = wsm[head * KW + k];

        // Lane t owns columns [7t, 7t+6]; 13-float window, halo reads hit the zero pads.
        const float* row = &tiles[wave][cur][LPAD];
        const int cbase = lane * 7;

        float xv[13];
#pragma unroll
        for (int i = 0; i < 13; ++i) xv[i] = row[cbase - PADW + i];   // guard-free

        float* orow = out + (size_t)r * WDIM + cbase;
#pragma unroll
        for (int c = 0; c < 7; ++c) {
            float acc = 0.f;
#pragma unroll
            for (int k = 0; k < KW; ++k) acc = fmaf(xv[c + k], wk[k], acc);
            __builtin_nontemporal_store(acc, orow + c);
        }
    }
}

extern "C" void kernel_launch(void* const* d_in, const int* in_sizes, int n_in,
                              void* d_out, int out_size, void* d_ws, size_t ws_size,
                              hipStream_t stream)
{
    (void)in_sizes; (void)n_in; (void)out_size; (void)d_ws; (void)ws_size;
    const float* x      = (const float*)d_in[0];
    const float* weight = (const float*)d_in[1];
    float*       out    = (float*)d_out;

    dim3 grid(NBLOCKS);
    dim3 block(WAVES_PER_BLOCK * 32);
    hipLaunchKernelGGL(lwconv2d_kernel, grid, block, 0, stream, x, weight, out);
}